// GNN_GCNConv_homogen_46153718563498
// MI455X (gfx1250) — compile-verified
//
#include <hip/hip_runtime.h>
#include <hip/hip_bf16.h>

typedef __attribute__((ext_vector_type(2))) float v2f;
typedef __attribute__((ext_vector_type(8))) float v8f;

#define N_NODES 100000
#define E_POS   500000
#define E_PRED  200000

// ---------------------------------------------------------------------------
// Degree / normalization kernels
// ---------------------------------------------------------------------------
__global__ void deg_init_kernel(float* deg, int n) {
    int i = blockIdx.x * blockDim.x + threadIdx.x;
    if (i < n) deg[i] = 1.0f;   // self-loop contributes 1
}

__global__ void deg_count_kernel(const int* __restrict__ tgt, float* deg, int e) {
    int i = blockIdx.x * blockDim.x + threadIdx.x;
    if (i < e) unsafeAtomicAdd(&deg[tgt[i]], 1.0f);
}

__global__ void deg_finalize_kernel(float* deg, int n) {
    int i = blockIdx.x * blockDim.x + threadIdx.x;
    if (i < n) deg[i] = rsqrtf(deg[i]);   // dinv = deg^-0.5
}

// ---------------------------------------------------------------------------
// WMMA GEMM: C[M,N] = A[M,K] @ B[K,N] (+ bias[N] if non-null)
// fp32 via V_WMMA_F32_16X16X4_F32. One wave (32 threads) per 16x16 tile.
// M % 16 == 0, N % 16 == 0, K % 4 == 0. EXEC all-ones (no divergence).
// ---------------------------------------------------------------------------
__global__ void gemm_wmma_f32(const float* __restrict__ A,
                              const float* __restrict__ B,
                              const float* __restrict__ bias,
                              float* __restrict__ C,
                              int K, int N) {
    const int m0   = blockIdx.x * 16;
    const int n0   = blockIdx.y * 16;
    const int lane = threadIdx.x;        // 0..31
    const int half = lane >> 4;          // 0 or 1
    const int idx  = lane & 15;

    v8f acc = {};
    const float* arow = A + (size_t)(m0 + idx) * K;   // A fragment row = lane&15
    const float* bcol = B + (size_t)(n0 + idx);       // B fragment col = lane&15

    for (int k0 = 0; k0 < K; k0 += 4) {
        const int kb = k0 + half * 2;    // K-pair per half-wave: {0,1} / {2,3}
        v2f a;
        a.x = arow[kb];
        a.y = arow[kb + 1];
        v2f b;
        b.x = bcol[(size_t)kb * N];
        b.y = bcol[(size_t)(kb + 1) * N];
        // (neg_a, A, neg_b, B, c_mod, C, reuse_a, reuse_b)
        acc = __builtin_amdgcn_wmma_f32_16x16x4_f32(
            false, a, false, b, (short)0, acc, false, false);
    }

    const float bv = bias ? bias[n0 + idx] : 0.0f;
    // C/D layout: VGPR j -> M = m0 + j + 8*half, N = n0 + (lane&15)
    float* crow = C + (size_t)(m0 + half * 8) * N + n0 + idx;
#pragma unroll
    for (int j = 0; j < 8; ++j) {
        crow[(size_t)j * N] = acc[j] + bv;
    }
}

// ---------------------------------------------------------------------------
// Edge scatter: agg[tgt, f] += xw[src, f] * dinv[src] * dinv[tgt]
// One thread per (edge, feature); F is a power of two (128 or 64).
// ---------------------------------------------------------------------------
__global__ void scatter_edges_kernel(const float* __restrict__ xw,
                                     const int* __restrict__ src,
                                     const int* __restrict__ tgt,
                                     const float* __restrict__ dinv,
                                     float* __restrict__ agg,
                                     int e, int f_log2) {
    long long i = (long long)blockIdx.x * blockDim.x + threadIdx.x;
    long long total = (long long)e << f_log2;
    if (i >= total) return;
    const int ed = (int)(i >> f_log2);
    const int f  = (int)(i & ((1 << f_log2) - 1));
    const int s  = src[ed];
    const int t  = tgt[ed];
    const float norm = dinv[s] * dinv[t];
    const float v = xw[((size_t)s << f_log2) + f] * norm;
    unsafeAtomicAdd(&agg[((size_t)t << f_log2) + f], v);
}

// ---------------------------------------------------------------------------
// Combine: out = (agg + xw * dinv^2 + bias[f]), optional relu
// ---------------------------------------------------------------------------
__global__ void combine_kernel(const float* __restrict__ agg,
                               const float* __restrict__ xw,
                               const float* __restrict__ dinv,
                               const float* __restrict__ bias,
                               float* __restrict__ out,
                               int n, int f_log2, int do_relu) {
    long long i = (long long)blockIdx.x * blockDim.x + threadIdx.x;
    long long total = (long long)n << f_log2;
    if (i >= total) return;
    const int node = (int)(i >> f_log2);
    const int f    = (int)(i & ((1 << f_log2) - 1));
    const float di = dinv[node];
    float v = agg[i] + xw[i] * di * di + bias[f];
    if (do_relu) v = fmaxf(v, 0.0f);
    out[i] = v;
}

// ---------------------------------------------------------------------------
// Edge scoring: logits[e] = dot(h[a_e], h[b_e]) over 64 dims.
// 16 lanes per edge, float4 per lane, shuffle reduction.
// ---------------------------------------------------------------------------
__global__ void edge_score_kernel(const float* __restrict__ h,
                                  const int* __restrict__ ia,
                                  const int* __restrict__ ib,
                                  float* __restrict__ out, int e) {
    int tid  = blockIdx.x * blockDim.x + threadIdx.x;
    int ed   = tid >> 4;
    int lane = tid & 15;
    if (ed >= e) return;
    const int a = ia[ed];
    const int b = ib[ed];
    const float4 va = ((const float4*)(h + (size_t)a * 64))[lane];
    const float4 vb = ((const float4*)(h + (size_t)b * 64))[lane];
    float s = va.x * vb.x + va.y * vb.y + va.z * vb.z + va.w * vb.w;
    s += __shfl_xor(s, 8, 32);
    s += __shfl_xor(s, 4, 32);
    s += __shfl_xor(s, 2, 32);
    s += __shfl_xor(s, 1, 32);
    if (lane == 0) out[ed] = s;
}

// ---------------------------------------------------------------------------
// Launch
// ---------------------------------------------------------------------------
extern "C" void kernel_launch(void* const* d_in, const int* in_sizes, int n_in,
                              void* d_out, int out_size, void* d_ws, size_t ws_size,
                              hipStream_t stream) {
    const float* x      = (const float*)d_in[0];
    const int*   eidx   = (const int*)d_in[1];   // [2, 200000]
    const int*   pidx   = (const int*)d_in[2];   // [2, 500000]
    const float* W_init = (const float*)d_in[3];
    const float* b_init = (const float*)d_in[4];
    const float* W1     = (const float*)d_in[5];
    const float* b1     = (const float*)d_in[6];
    const float* W2     = (const float*)d_in[7];
    const float* b2     = (const float*)d_in[8];
    float* logits = (float*)d_out;

    const int* src = pidx;             // pos_edge_index[0]
    const int* tgt = pidx + E_POS;     // pos_edge_index[1]
    const int* qa  = eidx;             // edge_index[0]
    const int* qb  = eidx + E_PRED;    // edge_index[1]

    // Workspace layout (floats)
    float* ws   = (float*)d_ws;
    float* dinv = ws;                                   // [100000]
    float* bufA = ws + 102400;                          // [N,128] h
    float* bufB = bufA + (size_t)N_NODES * 128;         // [N,128] xw
    float* bufC = bufB + (size_t)N_NODES * 128;         // [N,128] agg

    const int TB = 256;
    const int MT = N_NODES / 16;       // 6250 (exact)

    // --- degrees / dinv ---
    deg_init_kernel<<<(N_NODES + TB - 1) / TB, TB, 0, stream>>>(dinv, N_NODES);
    deg_count_kernel<<<(E_POS + TB - 1) / TB, TB, 0, stream>>>(tgt, dinv, E_POS);
    deg_finalize_kernel<<<(N_NODES + TB - 1) / TB, TB, 0, stream>>>(dinv, N_NODES);

    // --- h0 = x @ W_init + b_init -> bufA ---
    gemm_wmma_f32<<<dim3(MT, 128 / 16), 32, 0, stream>>>(x, W_init, b_init, bufA, 128, 128);

    // --- conv1: xw1 = h0 @ W1 -> bufB ---
    gemm_wmma_f32<<<dim3(MT, 128 / 16), 32, 0, stream>>>(bufA, W1, nullptr, bufB, 128, 128);
    hipMemsetAsync(bufC, 0, (size_t)N_NODES * 128 * sizeof(float), stream);
    {
        long long total = (long long)E_POS << 7;
        scatter_edges_kernel<<<(unsigned)((total + TB - 1) / TB), TB, 0, stream>>>(
            bufB, src, tgt, dinv, bufC, E_POS, 7);
        long long nf = (long long)N_NODES << 7;
        combine_kernel<<<(unsigned)((nf + TB - 1) / TB), TB, 0, stream>>>(
            bufC, bufB, dinv, b1, bufA, N_NODES, 7, 1);   // relu -> h1 in bufA
    }

    // --- conv2: xw2 = h1 @ W2 -> bufB (64 cols) ---
    gemm_wmma_f32<<<dim3(MT, 64 / 16), 32, 0, stream>>>(bufA, W2, nullptr, bufB, 128, 64);
    hipMemsetAsync(bufC, 0, (size_t)N_NODES * 64 * sizeof(float), stream);
    {
        long long total = (long long)E_POS << 6;
        scatter_edges_kernel<<<(unsigned)((total + TB - 1) / TB), TB, 0, stream>>>(
            bufB, src, tgt, dinv, bufC, E_POS, 6);
        long long nf = (long long)N_NODES << 6;
        combine_kernel<<<(unsigned)((nf + TB - 1) / TB), TB, 0, stream>>>(
            bufC, bufB, dinv, b2, bufA, N_NODES, 6, 0);   // h2 in bufA
    }

    // --- edge scoring ---
    {
        long long threads = (long long)E_PRED * 16;
        edge_score_kernel<<<(unsigned)((threads + TB - 1) / TB), TB, 0, stream>>>(
            bufA, qa, qb, logits, E_PRED);
    }
}